// ActorNet_4114578669786
// MI455X (gfx1250) — compile-verified
//
#include <hip/hip_runtime.h>

typedef __attribute__((ext_vector_type(2))) float v2f;
typedef __attribute__((ext_vector_type(8))) float v8f;
typedef __attribute__((ext_vector_type(4))) unsigned int u32x4;
typedef __attribute__((ext_vector_type(8))) int i32x8;
typedef __attribute__((ext_vector_type(4))) int i32x4;

#define B_SZ 1024
#define T_SZ 256
#define F_SZ 124
#define H_SZ 64
#define IN_SZ 128  // F + ADIM
#define G_SZ 256   // 4*H

#if defined(__has_builtin)
#if __has_builtin(__builtin_amdgcn_tensor_load_to_lds) && \
    __has_builtin(__builtin_amdgcn_s_wait_tensorcnt)
#define USE_TDM 1
#endif
#endif
#ifndef USE_TDM
#define USE_TDM 0
#endif

#if USE_TDM
// Build a D# (groups 0/1 per CDNA5 ISA) for a 2D tile of
// f32 and issue TENSOR_LOAD_TO_LDS. All args wave-uniform -> SGPR groups.
// 6-operand builtin on this toolchain:
//   (u32x4 group0, i32x8 group1, i32x4 group2, i32x4 group3, i32x8, i32 cpol)
static __device__ __forceinline__ void tdm_load_2d(
    unsigned lds_off, const void* gptr, unsigned tensor_d0, unsigned tensor_d1,
    unsigned tile_d0, unsigned tile_d1, unsigned long long stride0) {
  unsigned long long ga = (unsigned long long)(size_t)gptr;
  u32x4 g0;
  g0.x = 1u;                                  // count=1 (valid), user mode
  g0.y = lds_off;                             // LDS byte address [63:32]
  g0.z = (unsigned)(ga & 0xffffffffull);      // global_addr[31:0]  -> bits 95:64
  g0.w = (unsigned)((ga >> 32) & 0x01ffffffull)  // global_addr[56:32] -> 120:96
         | (2u << 30);                        // type=2 ("image")
  i32x8 g1;
  g1[0] = (int)(2u << 16);  // workgroup_mask=0 (no cluster), data_size=2 (4 B)
  g1[1] = (int)((tensor_d0 & 0xffffu) << 16);            // dim0[15:0]
  g1[2] = (int)((tensor_d0 >> 16) | ((tensor_d1 & 0xffffu) << 16));
  g1[3] = (int)((tensor_d1 >> 16) | (tile_d0 << 16));    // tile_dim0
  g1[4] = (int)(tile_d1 & 0xffffu);                      // tile_dim1, tile_dim2=0
  g1[5] = (int)(stride0 & 0xffffffffull);                // dim0_stride[31:0]
  g1[6] = (int)((stride0 >> 32) & 0xffffull);            // dim0_stride[47:32]
  g1[7] = 0;
  i32x4 z4 = {0, 0, 0, 0};          // groups 2/3 unused (<=2D tensor)
  i32x8 z8 = {0, 0, 0, 0, 0, 0, 0, 0};
  __builtin_amdgcn_tensor_load_to_lds(g0, g1, z4, z4, z8, 0);
}
#endif

static __device__ __forceinline__ v8f wmma4(v2f a, v2f b, v8f c) {
  // D = A(16x4,f32) * B(4x16,f32) + C(16x16,f32) -> v_wmma_f32_16x16x4_f32
  return __builtin_amdgcn_wmma_f32_16x16x4_f32(false, a, false, b, (short)0, c,
                                               false, false);
}

// One LSTM timestep. All LDS pointers are passed as constant-offset GEPs on the
// dynamic-shared base and the function is force-inlined, so every access stays
// addrspace(3) (ds_*), never flat. Ends after the fused-head accumulate; the
// caller places the tensor-wait + barrier D.
static __device__ __forceinline__ void lstm_step(
    const float* sXp, const float* sPap, const float* sW1p, const float* sWx,
    const float* sWh, float* sH, float* sG, float (&creg)[4],
    const float (&bi)[4], const float (&bf)[4], const float (&bg)[4],
    const float (&bo)[4], v8f& acc, int l16, int koff, int half, int n0,
    int wave, int erow, int ecol) {
  // ---- gates = [xs | h] (16x192) x [Wx;Wh] (192x256), our 32 columns ----
  v8f cg0 = {}, cg1 = {};
#pragma unroll
  for (int kk = 0; kk < IN_SZ; kk += 4) {
    int k = kk + koff;
    v2f a;
    if (kk < F_SZ) {  // folds at unroll time; 124 % 4 == 0 keeps chunks whole
      a.x = sXp[l16 * F_SZ + k];
      a.y = sXp[l16 * F_SZ + k + 1];
    } else {
      a.x = sPap[l16 * 4 + (k - F_SZ)];
      a.y = sPap[l16 * 4 + (k - F_SZ) + 1];
    }
    v2f bf0; bf0.x = sWx[k * G_SZ + n0 + l16];
             bf0.y = sWx[(k + 1) * G_SZ + n0 + l16];
    v2f bf1; bf1.x = sWx[k * G_SZ + n0 + 16 + l16];
             bf1.y = sWx[(k + 1) * G_SZ + n0 + 16 + l16];
    cg0 = wmma4(a, bf0, cg0);
    cg1 = wmma4(a, bf1, cg1);
  }
#pragma unroll
  for (int kk = 0; kk < H_SZ; kk += 4) {
    int k = kk + koff;
    v2f a;   a.x = sH[l16 * H_SZ + k];         a.y = sH[l16 * H_SZ + k + 1];
    v2f bf0; bf0.x = sWh[k * G_SZ + n0 + l16];
             bf0.y = sWh[(k + 1) * G_SZ + n0 + l16];
    v2f bf1; bf1.x = sWh[k * G_SZ + n0 + 16 + l16];
             bf1.y = sWh[(k + 1) * G_SZ + n0 + 16 + l16];
    cg0 = wmma4(a, bf0, cg0);
    cg1 = wmma4(a, bf1, cg1);
  }
#pragma unroll
  for (int v = 0; v < 8; ++v) {  // spill C frags -> sG for cross-wave use
    int row = v + half * 8;
    sG[row * G_SZ + n0 + l16]      = cg0[v];
    sG[row * G_SZ + n0 + 16 + l16] = cg1[v];
  }
  __syncthreads();  // barrier B: all 256 gate columns in LDS

  // ---- LSTM cell update: i,f,g,o -> c,h (4 elements / thread) ----
#pragma unroll
  for (int j = 0; j < 4; ++j) {
    int col = ecol + j;
    float pi = sG[erow * G_SZ + col]       + bi[j];
    float pf = sG[erow * G_SZ + 64 + col]  + bf[j];
    float pg = sG[erow * G_SZ + 128 + col] + bg[j];
    float po = sG[erow * G_SZ + 192 + col] + bo[j];
    float iv = 1.f / (1.f + __expf(-pi));
    float fv = 1.f / (1.f + __expf(-pf));
    float gv = tanhf(pg);
    float ov = 1.f / (1.f + __expf(-po));
    float cv = fv * creg[j] + iv * gv;
    creg[j] = cv;
    sH[erow * H_SZ + col] = ov * tanhf(cv);
  }
  __syncthreads();  // barrier C: sH(t) ready

  // ---- fused head: acc += h_t @ W1[t*64:(t+1)*64, :]  (waves 0..3) ----
  if (wave < 4) {
    const int na = wave * 16;
#pragma unroll
    for (int kk = 0; kk < H_SZ; kk += 4) {
      int k = kk + koff;
      v2f a;  a.x = sH[l16 * H_SZ + k];  a.y = sH[l16 * H_SZ + k + 1];
      v2f bw; bw.x = sW1p[k * H_SZ + na + l16];
              bw.y = sW1p[(k + 1) * H_SZ + na + l16];
      acc = wmma4(a, bw, acc);
    }
  }
}

__global__ __launch_bounds__(256) void actor_lstm_fused(
    const float* __restrict__ x, const float* __restrict__ pa,
    const float* __restrict__ Wx, const float* __restrict__ Wh,
    const float* __restrict__ bias, const float* __restrict__ W1,
    const float* __restrict__ b1, const float* __restrict__ W2,
    const float* __restrict__ b2, const float* __restrict__ W3,
    const float* __restrict__ b3, float* __restrict__ out_actor,
    float* __restrict__ out_c, float* __restrict__ out_h) {
  extern __shared__ float smem[];
  float* sWx  = smem;                   // 128*256 = 32768 f32 (128 KB)
  float* sWh  = sWx + IN_SZ * G_SZ;     // 64*256  = 16384 f32 ( 64 KB)
  float* sX0  = sWh + H_SZ * G_SZ;      // x tile [16 x 124], buffer 0
  float* sX1  = sX0 + 16 * F_SZ;        // x tile, buffer 1
  float* sPa0 = sX1 + 16 * F_SZ;        // pa tile [16 x 4], buffer 0
  float* sPa1 = sPa0 + 16 * 4;          // pa tile, buffer 1
  float* sW10 = sPa1 + 16 * 4;          // W1 tile [64 x 64], buffer 0
  float* sW11 = sW10 + H_SZ * H_SZ;     // W1 tile, buffer 1
  float* sH   = sW11 + H_SZ * H_SZ;     // 16*64
  float* sG   = sH + 16 * H_SZ;         // 16*256

  const int tid  = threadIdx.x;
  const int lane = tid & 31;
  const int wave = tid >> 5;
  const int half = lane >> 4;  // lanes 16..31 carry K+2 (A/B) / M+8 (C)
  const int l16  = lane & 15;
  const int koff = half * 2;
  const int bb0  = blockIdx.x * 16;

#if USE_TDM
#define PREFETCH_TILES(TT, SXQ, SPAQ, SW1Q)                                    \
  do {                                                                         \
    if ((TT) + 1 < T_SZ && wave == 0) {                                        \
      tdm_load_2d((unsigned)(size_t)(SXQ),                                     \
                  x + ((size_t)bb0 * T_SZ + (TT) + 1) * F_SZ, F_SZ, 16u, F_SZ, \
                  16u, (unsigned long long)T_SZ * F_SZ);                       \
      tdm_load_2d((unsigned)(size_t)(SPAQ),                                    \
                  pa + ((size_t)bb0 * T_SZ + (TT) + 1) * 4, 4u, 16u, 4u, 16u,  \
                  (unsigned long long)T_SZ * 4);                               \
      tdm_load_2d((unsigned)(size_t)(SW1Q), W1 + (size_t)((TT) + 1) * 4096,    \
                  4096u, 1u, 4096u, 1u, 4096ull);                              \
    }                                                                          \
  } while (0)
#define WAIT_TILES()                                              \
  do {                                                            \
    if (wave == 0) __builtin_amdgcn_s_wait_tensorcnt((short)0);   \
  } while (0)
  if (wave == 0) {
    // one-time staging of the LDS-resident weights via the Tensor Data Mover
    tdm_load_2d((unsigned)(size_t)sWx, Wx, 32768u, 1u, 32768u, 1u, 32768ull);
    tdm_load_2d((unsigned)(size_t)sWh, Wh, 16384u, 1u, 16384u, 1u, 16384ull);
    // preload tiles for t = 0 into buffer 0
    tdm_load_2d((unsigned)(size_t)sX0, x + (size_t)bb0 * T_SZ * F_SZ, F_SZ,
                16u, F_SZ, 16u, (unsigned long long)T_SZ * F_SZ);
    tdm_load_2d((unsigned)(size_t)sPa0, pa + (size_t)bb0 * T_SZ * 4, 4u, 16u,
                4u, 16u, (unsigned long long)T_SZ * 4);
    tdm_load_2d((unsigned)(size_t)sW10, W1, 4096u, 1u, 4096u, 1u, 4096ull);
    __builtin_amdgcn_s_wait_tensorcnt((short)0);
  }
#else
#define PREFETCH_TILES(TT, SXQ, SPAQ, SW1Q)                                    \
  do {                                                                         \
    if ((TT) + 1 < T_SZ) {                                                     \
      for (int e = tid; e < 16 * F_SZ; e += 256) {                             \
        int r = e / F_SZ, c = e - r * F_SZ;                                    \
        (SXQ)[e] = x[((size_t)(bb0 + r) * T_SZ + (TT) + 1) * F_SZ + c];        \
      }                                                                        \
      for (int e = tid; e < 64; e += 256)                                      \
        (SPAQ)[e] =                                                            \
            pa[((size_t)(bb0 + (e >> 2)) * T_SZ + (TT) + 1) * 4 + (e & 3)];    \
      for (int i = tid; i < 1024; i += 256)                                    \
        ((float4*)(SW1Q))[i] =                                                 \
            ((const float4*)(W1 + (size_t)((TT) + 1) * 4096))[i];              \
    }                                                                          \
  } while (0)
#define WAIT_TILES() do { } while (0)
  {
    const float4* gWx = (const float4*)Wx;
    float4* lWx = (float4*)sWx;
    for (int i = tid; i < (IN_SZ * G_SZ) / 4; i += 256) lWx[i] = gWx[i];
    const float4* gWh = (const float4*)Wh;
    float4* lWh = (float4*)sWh;
    for (int i = tid; i < (H_SZ * G_SZ) / 4; i += 256) lWh[i] = gWh[i];
    for (int e = tid; e < 16 * F_SZ; e += 256) {
      int r = e / F_SZ, c = e - r * F_SZ;
      sX0[e] = x[((size_t)(bb0 + r) * T_SZ + 0) * F_SZ + c];
    }
    for (int e = tid; e < 64; e += 256)
      sPa0[e] = pa[((size_t)(bb0 + (e >> 2)) * T_SZ + 0) * 4 + (e & 3)];
    for (int i = tid; i < 1024; i += 256)
      ((float4*)sW10)[i] = ((const float4*)W1)[i];
  }
#endif
  for (int e = tid; e < 16 * H_SZ; e += 256) sH[e] = 0.f;

  // per-thread cell state: this thread owns elements e0..e0+3 of the 16x64 c
  const int e0 = tid * 4;
  const int erow = e0 >> 6;
  const int ecol = e0 & 63;
  float creg[4] = {0.f, 0.f, 0.f, 0.f};
  float bi[4], bf[4], bg[4], bo[4];
#pragma unroll
  for (int j = 0; j < 4; ++j) {
    bi[j] = bias[ecol + j];
    bf[j] = bias[64 + ecol + j];
    bg[j] = bias[128 + ecol + j];
    bo[j] = bias[192 + ecol + j];
  }

  v8f acc = {};              // waves 0..3: running hs @ W1 tile (16x16)
  const int n0 = wave * 32;  // this wave's 32 gate columns
  __syncthreads();           // weights + t=0 tiles resident

  // Unrolled by 2 so the double-buffer choice is a compile-time constant:
  // keeps every tile access in addrspace(3) (ds_*), never flat.
  for (int t = 0; t < T_SZ; t += 2) {
    // even step: consume buffer 0, DMA t+1 into buffer 1
    PREFETCH_TILES(t, sX1, sPa1, sW11);
    lstm_step(sX0, sPa0, sW10, sWx, sWh, sH, sG, creg, bi, bf, bg, bo, acc,
              l16, koff, half, n0, wave, erow, ecol);
    WAIT_TILES();
    __syncthreads();  // barrier D: tiles for t+1 resident, sH reads done

    // odd step: consume buffer 1, DMA t+2 into buffer 0
    PREFETCH_TILES(t + 1, sX0, sPa0, sW10);
    lstm_step(sX1, sPa1, sW11, sWx, sWh, sH, sG, creg, bi, bf, bg, bo, acc,
              l16, koff, half, n0, wave, erow, ecol);
    WAIT_TILES();
    __syncthreads();  // barrier D'
  }

  // ---- MLP head: a1 = relu(acc+b1); a2 = relu(a1@W2+b2); actor = a2@W3+b3
  if (wave < 4) {
    const int col = wave * 16 + l16;
#pragma unroll
    for (int v = 0; v < 8; ++v) {
      int row = v + half * 8;
      sG[row * 64 + col] = fmaxf(acc[v] + b1[col], 0.f);  // a1 in sG[16x64]
    }
  }
  __syncthreads();
  if (wave < 4) {
    const int col = wave * 16 + l16;
    v8f c2 = {};
#pragma unroll
    for (int kk = 0; kk < H_SZ; kk += 4) {
      int k = kk + koff;
      v2f a;  a.x = sG[l16 * 64 + k];  a.y = sG[l16 * 64 + k + 1];
      v2f bw; bw.x = W2[k * 64 + col]; bw.y = W2[(k + 1) * 64 + col];
      c2 = wmma4(a, bw, c2);
    }
#pragma unroll
    for (int v = 0; v < 8; ++v) {
      int row = v + half * 8;
      sW10[row * 64 + col] = fmaxf(c2[v] + b2[col], 0.f);  // a2
    }
  }
  __syncthreads();
  if (wave == 0) {
    v8f c3 = {};
#pragma unroll
    for (int kk = 0; kk < H_SZ; kk += 4) {
      int k = kk + koff;
      v2f a;  a.x = sW10[l16 * 64 + k]; a.y = sW10[l16 * 64 + k + 1];
      v2f bw;
      bw.x = (l16 < 8) ? W3[k * 8 + l16] : 0.f;  // W3 [64,8] zero-padded to N=16
      bw.y = (l16 < 8) ? W3[(k + 1) * 8 + l16] : 0.f;
      c3 = wmma4(a, bw, c3);
    }
#pragma unroll
    for (int v = 0; v < 8; ++v) {
      int row = v + half * 8;
      if (l16 < 8) out_actor[(size_t)(bb0 + row) * 8 + l16] = c3[v] + b3[l16];
    }
  }

  // ---- final (c, h) ----
#pragma unroll
  for (int j = 0; j < 4; ++j) {
    out_c[(size_t)(bb0 + erow) * 64 + ecol + j] = creg[j];
    out_h[(size_t)(bb0 + erow) * 64 + ecol + j] = sH[erow * H_SZ + ecol + j];
  }
}

extern "C" void kernel_launch(void* const* d_in, const int* in_sizes, int n_in,
                              void* d_out, int out_size, void* d_ws,
                              size_t ws_size, hipStream_t stream) {
  const float* x  = (const float*)d_in[0];
  const float* pa = (const float*)d_in[1];
  const float* Wx = (const float*)d_in[2];
  const float* Wh = (const float*)d_in[3];
  const float* b  = (const float*)d_in[4];
  const float* W1 = (const float*)d_in[5];
  const float* b1 = (const float*)d_in[6];
  const float* W2 = (const float*)d_in[7];
  const float* b2 = (const float*)d_in[8];
  const float* W3 = (const float*)d_in[9];
  const float* b3 = (const float*)d_in[10];

  float* out       = (float*)d_out;
  float* out_actor = out;                  // [1024, 8]
  float* out_c     = out + B_SZ * 8;       // [1024, 64]
  float* out_h     = out_c + B_SZ * H_SZ;  // [1024, 64]

  const size_t shmem =
      (size_t)(IN_SZ * G_SZ + H_SZ * G_SZ +        // Wx, Wh
               2 * (16 * F_SZ) + 2 * (16 * 4) +    // x/pa double buffers
               2 * (H_SZ * H_SZ) +                 // W1 double buffer
               16 * H_SZ + 16 * G_SZ) *            // h, gates
      sizeof(float);  // ~260 KB; CDNA5 WGP LDS allows up to 320 KB/workgroup
  (void)hipFuncSetAttribute((const void*)actor_lstm_fused,
                            hipFuncAttributeMaxDynamicSharedMemorySize,
                            (int)shmem);

  actor_lstm_fused<<<dim3(B_SZ / 16), dim3(256), shmem, stream>>>(
      x, pa, Wx, Wh, b, W1, b1, W2, b2, W3, b3, out_actor, out_c, out_h);
}